// OhemCrossEntropy2d_15934328668326
// MI455X (gfx1250) — compile-verified
//
#include <hip/hip_runtime.h>
#include <hip/hip_bf16.h>
#include <stdint.h>

typedef unsigned int u32;
typedef __attribute__((ext_vector_type(4))) u32 u32x4;
typedef __attribute__((ext_vector_type(8))) int i32x8;
typedef __attribute__((ext_vector_type(4))) int i32x4;

#define NCLASS 19
#define HW (512 * 1024)          // pixels per image
#define TILE 256                 // pixels per TDM tile
#define TPI_SHIFT 11             // tiles per image = HW/TILE = 2048 = 1<<11
#define TPI_MASK 2047
#define TPB 8                    // tiles per block
#define MIN_KEPT_K 100000u
#define IGN 255
#define LOSS_BLOCKS 1024

// Order-preserving map: f1 < f2  <=>  fkey(f1) < fkey(f2) (totally ordered floats)
__device__ __forceinline__ u32 fkey(float f) {
  u32 b = __float_as_uint(f);
  return b ^ ((b & 0x80000000u) ? 0xFFFFFFFFu : 0x80000000u);
}

// Issue a TDM load of a 19 x 256 f32 tile (row stride = HW elements) into LDS.
// D# group0: count=1 | lds_addr | global_addr[56:0] | type=2
// D# group1: data_size=4B, tensor_dim0=1<<30 (no OOB), tensor_dim1=19,
//            tile_dim0=256, tile_dim1=19, tensor_dim0_stride=HW
__device__ __forceinline__ void tdm_issue(const float* gsrc_tile, void* lds_dst) {
  unsigned long long ga = (unsigned long long)(uintptr_t)gsrc_tile;
  u32 lds = (u32)(uintptr_t)lds_dst;  // low 32 bits of flat shared ptr = LDS byte addr
  u32x4 g0;
  g0.x = 1u;                                             // count=1, user descriptor
  g0.y = lds;                                            // lds_addr
  g0.z = (u32)ga;                                        // global_addr[31:0]
  g0.w = ((u32)(ga >> 32) & 0x01FFFFFFu) | 0x80000000u;  // global_addr[56:32] | type=2
  i32x8 g1;
  g1[0] = (int)(2u << 16);                               // data_size = 4 bytes
  g1[1] = 0;                                             // tensor_dim0 lo16 (td0=1<<30 -> 0)
  g1[2] = (int)(0x4000u | ((u32)NCLASS << 16));          // td0 hi16; tensor_dim1 lo16 = 19
  g1[3] = (int)((u32)TILE << 16);                        // td1 hi16 = 0; tile_dim0 = 256
  g1[4] = (int)NCLASS;                                   // tile_dim1 = 19; tile_dim2 = 0
  g1[5] = (int)HW;                                       // tensor_dim0_stride lo32
  g1[6] = 0;
  g1[7] = 0;
  i32x4 z4 = {0, 0, 0, 0};
#if __clang_major__ >= 23
  i32x8 z8 = {0, 0, 0, 0, 0, 0, 0, 0};
  __builtin_amdgcn_tensor_load_to_lds(g0, g1, z4, z4, z8, 0);
#else
  __builtin_amdgcn_tensor_load_to_lds(g0, g1, z4, z4, 0);
#endif
}

__device__ __forceinline__ const float* tile_ptr(const float* logits, int T) {
  int n = T >> TPI_SHIFT;      // image index (shift, no division)
  int t = T & TPI_MASK;        // tile index within image
  return logits + ((size_t)n * (NCLASS * (size_t)HW) + (size_t)t * TILE);
}

// Pass 1: TDM double-buffered streaming of channel tiles -> per-pixel log-softmax
// at label, store logp (NaN for ignored), fused level-1 radix histogram (key>>24).
__global__ __launch_bounds__(256) void k_pass1(const float* __restrict__ logits,
                                               const int* __restrict__ tgt,
                                               float* __restrict__ logp_out,
                                               u32* __restrict__ hist1) {
  __shared__ float buf[2][NCLASS * TILE];
  __shared__ u32 shist[256];
  shist[threadIdx.x] = 0;
  // SCALAR (SGPR) wave id => s_cmp/s_cbranch, so waves 1..7 truly branch AROUND
  // the TDM issue (TDM ignores EXEC; mask-predication would duplicate the DMA 8x).
  const int wave_sc = __builtin_amdgcn_readfirstlane((int)threadIdx.x) >> 5;
  const int tile0 = (int)blockIdx.x * TPB;  // TPI divisible by TPB: no image crossing

  if (wave_sc == 0) {
    tdm_issue(tile_ptr(logits, tile0), &buf[0][0]);
  }
  __syncthreads();  // also covers shist init

  for (int i = 0; i < TPB; ++i) {
    const int T = tile0 + i;
    if (wave_sc == 0) {
      if (i + 1 < TPB) {
        tdm_issue(tile_ptr(logits, T + 1), &buf[(i + 1) & 1][0]);
        __builtin_amdgcn_s_wait_tensorcnt(1);  // oldest (tile i) complete
      } else {
        __builtin_amdgcn_s_wait_tensorcnt(0);
      }
    }
    __syncthreads();

    const float* tb = &buf[i & 1][0];
    const int p = (T << 8) + (int)threadIdx.x;  // global pixel index (T*TILE + tid)

    int lab = tgt[p];
    bool valid = (lab != IGN);
    int sl = (valid && (u32)lab < (u32)NCLASS) ? lab : 0;

    float v[NCLASS];
    float m = -3.4e38f;
#pragma unroll
    for (int c = 0; c < NCLASS; ++c) {
      v[c] = tb[c * TILE + threadIdx.x];
      m = fmaxf(m, v[c]);
    }
    float s = 0.f;
#pragma unroll
    for (int c = 0; c < NCLASS; ++c) s += __expf(v[c] - m);
    float xl = tb[sl * TILE + threadIdx.x];  // dynamic index via LDS (no scratch spill)
    float logp = (xl - m) - __logf(s);

    logp_out[p] = valid ? logp : __uint_as_float(0x7FC00000u);  // NaN marks ignored
    if (valid) atomicAdd(&shist[fkey(logp) >> 24], 1u);
    __syncthreads();  // everyone done with buf[i&1] before it is refilled
  }
  u32 h = shist[threadIdx.x];
  if (h) atomicAdd(&hist1[threadIdx.x], h);  // integer atomics: order-independent
}

// Radix-select refinement: histogram next 8 key bits for keys matching the prefix.
__global__ __launch_bounds__(256) void k_refine(const float4* __restrict__ lp,
                                                const u32* __restrict__ ctrl,
                                                u32* __restrict__ histo, int n4,
                                                int matchShift, int binShift) {
  __shared__ u32 sh[256];
  sh[threadIdx.x] = 0;
  __syncthreads();
  const u32 kpref = ctrl[0];
  for (int i = (int)(blockIdx.x * blockDim.x + threadIdx.x); i < n4;
       i += (int)(gridDim.x * blockDim.x)) {
    float4 q = lp[i];
    float vv[4] = {q.x, q.y, q.z, q.w};
#pragma unroll
    for (int j = 0; j < 4; ++j) {
      u32 k = fkey(vv[j]);  // NaN keys have top byte 0xFF -> never match a valid prefix
      if ((k >> matchShift) == kpref) atomicAdd(&sh[(k >> binShift) & 0xFFu], 1u);
    }
  }
  __syncthreads();
  u32 h = sh[threadIdx.x];
  if (h) atomicAdd(&histo[threadIdx.x], h);
}

// Tiny serial scan over 256 bins: advance prefix/rank; final level writes threshold.
__global__ void k_scan(const u32* __restrict__ hist, u32* __restrict__ ctrl,
                       int first, int last) {
  if (threadIdx.x != 0 || blockIdx.x != 0) return;
  u32 rank = first ? MIN_KEPT_K : ctrl[1];
  u32 kpref = first ? 0u : ctrl[0];
  if (first) {
    u32 tot = 0;
    for (int i = 0; i < 256; ++i) tot += hist[i];
    ctrl[2] = tot;  // num_valid (ignored pixels never histogrammed)
  }
  u32 cum = 0, b = 255;
  for (int i = 0; i < 256; ++i) {
    u32 h = hist[i];
    if (cum + h >= rank) { b = (u32)i; rank -= cum; break; }
    cum += h;
  }
  kpref = (kpref << 8) | b;
  ctrl[0] = kpref;
  ctrl[1] = rank;
  if (last) {
    u32 bits = (kpref & 0x80000000u) ? (kpref ^ 0x80000000u) : ~kpref;
    float kth_logp = __uint_as_float(bits);
    float thr = fmaxf(kth_logp, logf(0.7f));  // max(kth, 0.7) in log domain
    if (ctrl[2] <= MIN_KEPT_K) thr = __uint_as_float(0x7F800000u);  // keep all valid
    ctrl[3] = __float_as_uint(thr);
  }
}

// Loss partials: wave32 shuffle reduce + per-block slot (deterministic, no float atomics).
__global__ __launch_bounds__(256) void k_loss(const float4* __restrict__ lp,
                                              const u32* __restrict__ ctrl,
                                              float* __restrict__ psum,
                                              u32* __restrict__ pcnt, int n4) {
  __shared__ float ws[8];
  __shared__ u32 wc[8];
  const float thr = __uint_as_float(ctrl[3]);
  float s = 0.f;
  u32 c = 0;
  for (int i = (int)(blockIdx.x * blockDim.x + threadIdx.x); i < n4;
       i += (int)(gridDim.x * blockDim.x)) {
    float4 q = lp[i];  // NaN (ignored) fails all compares -> excluded
    if (q.x <= thr) { s -= q.x; ++c; }
    if (q.y <= thr) { s -= q.y; ++c; }
    if (q.z <= thr) { s -= q.z; ++c; }
    if (q.w <= thr) { s -= q.w; ++c; }
  }
#pragma unroll
  for (int o = 16; o > 0; o >>= 1) {  // wave32 reduction
    s += __shfl_down(s, o, 32);
    c += __shfl_down(c, o, 32);
  }
  const int wave = (int)(threadIdx.x >> 5);
  if ((threadIdx.x & 31) == 0) { ws[wave] = s; wc[wave] = c; }
  __syncthreads();
  if (threadIdx.x == 0) {
    float bs = 0.f;
    u32 bc = 0;
#pragma unroll
    for (int w = 0; w < 8; ++w) { bs += ws[w]; bc += wc[w]; }
    psum[blockIdx.x] = bs;  // fixed slot: deterministic
    pcnt[blockIdx.x] = bc;
  }
}

// Deterministic fixed-order final reduction of per-block partials.
__global__ __launch_bounds__(256) void k_finish(const float* __restrict__ psum,
                                                const u32* __restrict__ pcnt,
                                                float* __restrict__ out) {
  __shared__ float ss[256];
  __shared__ u32 sc[256];
  float s = 0.f;
  u32 c = 0;
  for (int i = (int)threadIdx.x; i < LOSS_BLOCKS; i += 256) {
    s += psum[i];
    c += pcnt[i];
  }
  ss[threadIdx.x] = s;
  sc[threadIdx.x] = c;
  __syncthreads();
#pragma unroll
  for (int o = 128; o > 0; o >>= 1) {
    if ((int)threadIdx.x < o) {
      ss[threadIdx.x] += ss[threadIdx.x + o];
      sc[threadIdx.x] += sc[threadIdx.x + o];
    }
    __syncthreads();
  }
  if (threadIdx.x == 0) {
    u32 cc = sc[0];
    out[0] = ss[0] / (float)(cc ? cc : 1u);
  }
}

extern "C" void kernel_launch(void* const* d_in, const int* in_sizes, int n_in,
                              void* d_out, int out_size, void* d_ws, size_t ws_size,
                              hipStream_t stream) {
  const float* logits = (const float*)d_in[0];
  const int* tgt = (const int*)d_in[1];
  float* out = (float*)d_out;

  const long long P = (long long)in_sizes[1];        // 8*512*1024 pixels
  const int blocks1 = (int)((P / TILE) / TPB);       // 2048

  // Workspace: P f32 logp values | 4x256 hist bins | ctrl | per-block partials.
  float* ws_logp = (float*)d_ws;
  u32* hist = (u32*)((char*)d_ws + (size_t)P * sizeof(float));
  u32* h1 = hist;
  u32* h2 = hist + 256;
  u32* h3 = hist + 512;
  u32* h4 = hist + 768;
  u32* ctrl = hist + 1024;  // [0]=prefix [1]=rank [2]=num_valid [3]=thr bits
  float* psum = (float*)(ctrl + 16);
  u32* pcnt = (u32*)(psum + LOSS_BLOCKS);
  hipMemsetAsync((void*)hist, 0, (1024 + 16) * sizeof(u32), stream);

  k_pass1<<<blocks1, 256, 0, stream>>>(logits, tgt, ws_logp, h1);
  const int n4 = (int)(P / 4);
  k_scan<<<1, 1, 0, stream>>>(h1, ctrl, 1, 0);
  k_refine<<<LOSS_BLOCKS, 256, 0, stream>>>((const float4*)ws_logp, ctrl, h2, n4, 24, 16);
  k_scan<<<1, 1, 0, stream>>>(h2, ctrl, 0, 0);
  k_refine<<<LOSS_BLOCKS, 256, 0, stream>>>((const float4*)ws_logp, ctrl, h3, n4, 16, 8);
  k_scan<<<1, 1, 0, stream>>>(h3, ctrl, 0, 0);
  k_refine<<<LOSS_BLOCKS, 256, 0, stream>>>((const float4*)ws_logp, ctrl, h4, n4, 8, 0);
  k_scan<<<1, 1, 0, stream>>>(h4, ctrl, 0, 1);
  k_loss<<<LOSS_BLOCKS, 256, 0, stream>>>((const float4*)ws_logp, ctrl, psum, pcnt, n4);
  k_finish<<<1, 256, 0, stream>>>(psum, pcnt, out);
}